// PCEN_63857573757498
// MI455X (gfx1250) — compile-verified
//
#include <hip/hip_runtime.h>
#include <hip/hip_bf16.h>
#include <stdint.h>

// Problem geometry (fixed by setup_inputs)
#define B_    128
#define T_    2048
#define F_    128
#define L_    128                 // timesteps per chunk
#define NC_   (T_ / L_)           // 16 chunks
#define SUB_  32                  // timesteps per LDS sub-tile
#define NSUB_ (L_ / SUB_)         // 4 sub-tiles per chunk
#define SUB_BYTES (SUB_ * F_ * 4) // 16 KiB, globally contiguous
#define EPS_  1e-6f

// ---------------- CDNA5 async global->LDS path ----------------------------
#if __has_builtin(__builtin_amdgcn_global_load_async_to_lds_b128)
#define HAVE_ASYNC_BUILTIN 1
#else
#define HAVE_ASYNC_BUILTIN 0
#endif

typedef int v4i_t __attribute__((vector_size(16)));
typedef __attribute__((address_space(1))) v4i_t gm_v4i;   // global (as1)
typedef __attribute__((address_space(3))) v4i_t lds_v4i;  // LDS (as3)

__device__ __forceinline__ void issue_async_b128(const void* g, void* l) {
#if HAVE_ASYNC_BUILTIN
  __builtin_amdgcn_global_load_async_to_lds_b128(
      (gm_v4i*)(uintptr_t)g,
      (lds_v4i*)(uint32_t)(uintptr_t)l, /*offset=*/0, /*cpol=*/0);
#else
  // GV-mode VGLOBAL async load: VDST = per-lane LDS byte address,
  // VADDR = 64-bit global address (CDNA5 ISA).
  uint64_t ga = (uint64_t)(uintptr_t)g;
  uint32_t la = (uint32_t)(uintptr_t)l;  // low 32 bits of generic LDS ptr = LDS offset
  asm volatile("global_load_async_to_lds_b128 %0, %1, off"
               :: "v"(la), "v"(ga) : "memory");
#endif
}

#if __has_builtin(__builtin_amdgcn_s_wait_asynccnt)
#define WAIT_ASYNC(n) __builtin_amdgcn_s_wait_asynccnt(n)
#else
#define WAIT_ASYNC(n) asm volatile("s_wait_asynccnt %0" :: "i"(n) : "memory")
#endif

// Raw hardware exp2 (v_exp_f32); args here are in [-1, 20] -> safe.
#if __has_builtin(__builtin_amdgcn_exp2f)
#define EXP2F(x) __builtin_amdgcn_exp2f(x)
#else
#define EXP2F(x) exp2f(x)
#endif

// Copy one contiguous 16 KiB sub-tile (32 timesteps x 128 feats of f32) into LDS.
// 128 threads x 16 B = 2 KiB per issue -> 8 async b128 issues per wave.
__device__ __forceinline__ void stage_subtile(const char* gsrc, char* lbuf, int tid) {
#pragma unroll
  for (int j = 0; j < SUB_BYTES / (F_ * 16); ++j) {  // 8
    const int off = (j * F_ + tid) * 16;
    issue_async_b128(gsrc + off, lbuf + off);
  }
}

// ---------------- Pass 1: chunk-local terminal energies --------------------
// grid = B*NC blocks, 128 threads (one feature per lane; 4 wave32/block)
__global__ __launch_bounds__(F_) void pcen_chunk_local(
    const float* __restrict__ x, const float* __restrict__ log_s,
    float* __restrict__ ws) {
  __shared__ __align__(16) float lbuf[2][SUB_ * F_];  // 32 KiB double buffer
  const int tid = threadIdx.x;
  const int b = blockIdx.x / NC_;
  const int c = blockIdx.x % NC_;
  const float s = __expf(log_s[tid]);
  const float w = 1.0f - s;
  const char* gx = (const char*)(x + ((size_t)b * T_ + (size_t)c * L_) * F_);

  stage_subtile(gx, (char*)lbuf[0], tid);
  stage_subtile(gx + SUB_BYTES, (char*)lbuf[1], tid);

  float e = 0.0f;  // chunk-local: homogeneous part handled in pass 2/3
#pragma unroll
  for (int st = 0; st < NSUB_; ++st) {
    if (st < NSUB_ - 1) { WAIT_ASYNC(8); } else { WAIT_ASYNC(0); }
    __syncthreads();  // current buffer complete for all waves
    const float* rb = lbuf[st & 1];
#pragma unroll
    for (int t = 0; t < SUB_; ++t)
      e = fmaf(w, e, s * rb[t * F_ + tid]);
    __syncthreads();  // everyone done reading before this buffer is re-staged
    if (st + 2 < NSUB_)
      stage_subtile(gx + (size_t)(st + 2) * SUB_BYTES, (char*)lbuf[st & 1], tid);
  }
  ws[((size_t)b * NC_ + c) * F_ + tid] = e;
}

// ---------------- Pass 2: serial combine across chunks ---------------------
// grid = B blocks, 128 threads; ws[b][c][f] becomes the carry-in E[c*L-1]
__global__ __launch_bounds__(F_) void pcen_combine(
    const float* __restrict__ state, const float* __restrict__ log_s,
    float* __restrict__ ws) {
  const int f = threadIdx.x;
  const int b = blockIdx.x;
  const float s = __expf(log_s[f]);
  const float A = __expf((float)L_ * log1pf(-s));  // (1-s)^L
  float e = state[(size_t)b * F_ + f];             // E[-1]
#pragma unroll
  for (int c = 0; c < NC_; ++c) {
    const size_t idx = ((size_t)b * NC_ + c) * F_ + f;
    const float local_end = ws[idx];
    ws[idx] = e;                   // carry-in for chunk c
    e = fmaf(A, e, local_end);     // E at end of chunk c
  }
}

// ---------------- Pass 3: recompute E exactly + emit y, new_state ----------
__global__ __launch_bounds__(F_) void pcen_apply(
    const float* __restrict__ x, const float* __restrict__ log_s,
    const float* __restrict__ log_alpha, const float* __restrict__ log_delta,
    const float* __restrict__ log_r, const float* __restrict__ ws,
    float* __restrict__ out) {
  __shared__ __align__(16) float lbuf[2][SUB_ * F_];
  const int tid = threadIdx.x;
  const int b = blockIdx.x / NC_;
  const int c = blockIdx.x % NC_;
  const float s = __expf(log_s[tid]);
  const float w = 1.0f - s;
  const float alpha = __expf(log_alpha[tid]);
  const float delta = __expf(log_delta[tid]);
  const float r = __expf(log_r[tid]);
  const float dr = __expf(r * log_delta[tid]);  // delta^r = exp(r*log_delta)

  const size_t base_elt = ((size_t)b * T_ + (size_t)c * L_) * F_;
  const char* gx = (const char*)(x + base_elt);
  float* yout = out + base_elt + tid;

  stage_subtile(gx, (char*)lbuf[0], tid);
  stage_subtile(gx + SUB_BYTES, (char*)lbuf[1], tid);

  float e = ws[((size_t)b * NC_ + c) * F_ + tid];  // exact carry-in
#pragma unroll
  for (int st = 0; st < NSUB_; ++st) {
    if (st < NSUB_ - 1) { WAIT_ASYNC(8); } else { WAIT_ASYNC(0); }
    __syncthreads();
    const float* rb = lbuf[st & 1];
#pragma unroll
    for (int t = 0; t < SUB_; ++t) {
      const float xv = rb[t * F_ + tid];
      e = fmaf(w, e, s * xv);
      // (EPS+E)^-alpha and (...)^r via native v_log_f32/v_exp_f32 (log2/exp2)
      const float gain = EXP2F(-alpha * __log2f(e + EPS_));
      const float yv = EXP2F(r * __log2f(fmaf(xv, gain, delta))) - dr;
      // NT store: keep streaming y out of L2 so x stays resident for reuse
      __builtin_nontemporal_store(yv, yout + (size_t)(st * SUB_ + t) * F_);
    }
    __syncthreads();
    if (st + 2 < NSUB_)
      stage_subtile(gx + (size_t)(st + 2) * SUB_BYTES, (char*)lbuf[st & 1], tid);
  }
  if (c == NC_ - 1)  // new_state = E[:, -1, :]
    out[(size_t)B_ * T_ * F_ + (size_t)b * F_ + tid] = e;
}

// ---------------- Host wiring ----------------------------------------------
extern "C" void kernel_launch(void* const* d_in, const int* in_sizes, int n_in,
                              void* d_out, int out_size, void* d_ws, size_t ws_size,
                              hipStream_t stream) {
  const float* x         = (const float*)d_in[0];
  const float* state     = (const float*)d_in[1];
  const float* log_s     = (const float*)d_in[2];
  const float* log_alpha = (const float*)d_in[3];
  const float* log_delta = (const float*)d_in[4];
  const float* log_r     = (const float*)d_in[5];
  float* out = (float*)d_out;
  float* ws  = (float*)d_ws;  // B*NC*F floats = 1 MiB

  pcen_chunk_local<<<B_ * NC_, F_, 0, stream>>>(x, log_s, ws);
  pcen_combine<<<B_, F_, 0, stream>>>(state, log_s, ws);
  pcen_apply<<<B_ * NC_, F_, 0, stream>>>(x, log_s, log_alpha, log_delta,
                                          log_r, ws, out);
}